// DFTProcessor_46883863003254
// MI455X (gfx1250) — compile-verified
//
#include <hip/hip_runtime.h>

#define PI_F 3.14159265358979323846f

#define E_TOT   768
#define T_LEN   4096
#define N2      2048          // complex FFT length
#define NFREQ   2049          // rfft bins
#define FP      2052          // padded per-series pitch in complex elements
#define NSER    12288         // 16*768 forward series == 192*64 inverse series
#define PITCH   66            // LDS row pitch in floats (conflict-free, 8B-aligned)

typedef float v2f __attribute__((ext_vector_type(2)));
typedef float v8f __attribute__((ext_vector_type(8)));

// ---------------------------------------------------------------------------
// Kernel 1: forward real FFT (length 4096) per (b, e) series.
// Packs x[2n],x[2n+1] into 2048 complex, Stockham DIF radix-2 in LDS,
// unpacks to 2049 rfft bins, stores to ws layout [bm][em][f] (f contiguous).
// ---------------------------------------------------------------------------
__global__ __launch_bounds__(256) void fwd_rfft_kernel(
        const float* __restrict__ x, float2* __restrict__ spec)
{
    __shared__ float2 bufA[N2];
    __shared__ float2 bufB[N2];
    const int tid = threadIdx.x;
    const int s   = blockIdx.x;            // b*768 + e
    const int b   = s / E_TOT;
    const int e   = s - b * E_TOT;

    const float* xb = x + (size_t)b * T_LEN * E_TOT + e;
    for (int i = 0; i < 8; ++i) {
        int n = tid + i * 256;
        float re = xb[(size_t)(2 * n    ) * E_TOT];
        float im = xb[(size_t)(2 * n + 1) * E_TOT];
        bufA[n] = make_float2(re, im);
    }

    float2* src = bufA;
    float2* dst = bufB;
    int lm = 0, m = 1;
    for (int l = N2 / 2; l >= 1; l >>= 1) {
        __syncthreads();
        float angScale = -PI_F / (float)l;
        for (int i = 0; i < 4; ++i) {
            int t = tid + i * 256;               // 0..1023 butterflies
            int k = t & (m - 1);
            int j = t >> lm;
            float2 c0 = src[k + (j << lm)];
            float2 c1 = src[k + (j << lm) + N2 / 2];
            float2 su = make_float2(c0.x + c1.x, c0.y + c1.y);
            float2 di = make_float2(c0.x - c1.x, c0.y - c1.y);
            float sn, cs;
            __sincosf(angScale * (float)j, &sn, &cs);
            dst[k + (j << (lm + 1))]     = su;
            dst[k + (j << (lm + 1)) + m] = make_float2(cs * di.x - sn * di.y,
                                                       cs * di.y + sn * di.x);
        }
        float2* tp = src; src = dst; dst = tp;
        m <<= 1; ++lm;
    }
    __syncthreads();

    // real-FFT unpack: X[k] = E[k] + W^k O[k], W = exp(-2pi i /4096)
    const int bm = b * 12 + (e >> 6);
    const int em = e & 63;
    float2* out = spec + (size_t)(bm * 64 + em) * FP;
    for (int k = tid; k <= N2; k += 256) {
        float2 Zk = (k < N2) ? src[k] : src[0];
        float2 Zm = src[(N2 - k) & (N2 - 1)];
        Zm.y = -Zm.y;
        float2 Ev = make_float2(0.5f * (Zk.x + Zm.x), 0.5f * (Zk.y + Zm.y));
        float2 Dv = make_float2(Zk.x - Zm.x, Zk.y - Zm.y);
        float2 Ov = make_float2(0.5f * Dv.y, -0.5f * Dv.x);   // -0.5i * D
        float sn, cs;
        __sincosf(-PI_F * (float)k / 2048.0f, &sn, &cs);
        out[k] = make_float2(Ev.x + cs * Ov.x - sn * Ov.y,
                             Ev.y + cs * Ov.y + sn * Ov.x);
    }
}

// ---------------------------------------------------------------------------
// Kernel 2: per-frequency complex GEMMs with V_WMMA_F32_16X16X4_F32.
//   invden[n] = 1/(1 - A[n] * exp(-2pi i f/4096))
//   Bs[e][n]  = invden[n] * B[n][e]                (transposed, in LDS)
//   G[a][e]   = sum_n C[a][n] * Bs[e][n]           (phase A, WMMA)
//   Y[bm][a]  = sum_e X[bm][e] * G[a][e] + bias    (phase B, WMMA) -> cReLU
// Complex = 4 real accumulators: Re = RR - II, Im = RI + IR.
// LDS holds re/im in SEPARATE planes so each WMMA fragment is one ds_load_b64
// (no repack movs in the inner loop).
// ---------------------------------------------------------------------------
__global__ __launch_bounds__(256) void freq_matmul_kernel(
        float2* __restrict__ spec,
        const float* __restrict__ A_re, const float* __restrict__ A_im,
        const float* __restrict__ B_re, const float* __restrict__ B_im,
        const float* __restrict__ C_re, const float* __restrict__ C_im,
        const float* __restrict__ bias_re, const float* __restrict__ bias_im)
{
    extern __shared__ float lds[];
    float*  Cr  = lds;                   // [64][PITCH]  C[a][n]
    float*  Ci  = Cr  + 64 * PITCH;
    float*  Btr = Ci  + 64 * PITCH;      // [64][PITCH]  Bs[e][n]
    float*  Bti = Btr + 64 * PITCH;
    float*  Gr  = Bti + 64 * PITCH;      // [64][PITCH]  G[a][e]
    float*  Gi  = Gr  + 64 * PITCH;
    float*  Xr  = Gi  + 64 * PITCH;      // [192][PITCH] X[bm][e]
    float*  Xi  = Xr  + 192 * PITCH;
    float2* dnv = (float2*)(Xi + 192 * PITCH);   // [64] invden

    const int tid   = threadIdx.x;
    const int f     = blockIdx.x;
    const int lane  = tid & 31;          // wave32
    const int w     = tid >> 5;          // 8 waves
    const int l15   = lane & 15;
    const int kbase = (lane >> 4) << 1;  // 0: K=0,1   2: K=2,3

    // stage C (frequency independent)
    for (int i = 0; i < 16; ++i) {
        int idx = tid + i * 256;                 // 0..4095
        int a = idx >> 6, n = idx & 63;
        Cr[a * PITCH + n] = C_re[idx];
        Ci[a * PITCH + n] = C_im[idx];
    }
    // resolvent invden[n]
    if (tid < 64) {
        float th = PI_F * (float)f / 2048.0f;    // 2*pi*f/4096
        float sn, cs;
        __sincosf(th, &sn, &cs);                 // w = (cs, -sn)
        float ar = A_re[tid], ai = A_im[tid];
        float dr = 1.0f - (ar * cs + ai * sn);
        float di = -(ai * cs - ar * sn);
        float r2 = 1.0f / (dr * dr + di * di);
        dnv[tid] = make_float2(dr * r2, -di * r2);
    }
    __syncthreads();
    // Bs[e][n] = invden[n] * B[n][e]   (transpose into LDS)
    for (int i = 0; i < 16; ++i) {
        int idx = tid + i * 256;
        int n = idx >> 6, e = idx & 63;
        float br = B_re[idx], bi = B_im[idx];
        float2 d = dnv[n];
        Btr[e * PITCH + n] = br * d.x - bi * d.y;
        Bti[e * PITCH + n] = br * d.y + bi * d.x;
    }
    // stage X_f [192][64]
    for (int i = 0; i < 48; ++i) {
        int idx = tid + i * 256;                 // 0..12287 == bm*64+e
        int bm = idx >> 6, e = idx & 63;
        float2 v = spec[(size_t)idx * FP + f];
        Xr[bm * PITCH + e] = v.x;
        Xi[bm * PITCH + e] = v.y;
    }
    __syncthreads();

    // ---- phase A: G = C x Bs^T  (16 complex 16x16 tiles, 2 per wave) ----
    for (int p = 2 * w; p < 2 * w + 2; ++p) {
        int atile = p >> 2, etile = p & 3;
        v8f RR = {}, II = {}, RI = {}, IR = {};
        const float* ArR = Cr  + (atile * 16 + l15) * PITCH + kbase;
        const float* ArI = Ci  + (atile * 16 + l15) * PITCH + kbase;
        const float* BrR = Btr + (etile * 16 + l15) * PITCH + kbase;
        const float* BrI = Bti + (etile * 16 + l15) * PITCH + kbase;
#pragma unroll
        for (int kk = 0; kk < 16; ++kk) {
            v2f ar = *(const v2f*)(ArR + kk * 4);
            v2f ai = *(const v2f*)(ArI + kk * 4);
            v2f br = *(const v2f*)(BrR + kk * 4);
            v2f bi = *(const v2f*)(BrI + kk * 4);
            RR = __builtin_amdgcn_wmma_f32_16x16x4_f32(false, ar, false, br, (short)0, RR, false, false);
            II = __builtin_amdgcn_wmma_f32_16x16x4_f32(false, ai, false, bi, (short)0, II, false, false);
            RI = __builtin_amdgcn_wmma_f32_16x16x4_f32(false, ar, false, bi, (short)0, RI, false, false);
            IR = __builtin_amdgcn_wmma_f32_16x16x4_f32(false, ai, false, br, (short)0, IR, false, false);
        }
#pragma unroll
        for (int j = 0; j < 8; ++j) {
            int row = atile * 16 + ((lane < 16) ? j : j + 8);
            int col = etile * 16 + l15;
            Gr[row * PITCH + col] = RR[j] - II[j];
            Gi[row * PITCH + col] = RI[j] + IR[j];
        }
    }
    __syncthreads();

    // ---- phase B: Y = X x G^T + bias, complex ReLU (48 tiles, 6 per wave) ----
    const int atile = w & 3;
    const float brb = bias_re[atile * 16 + l15];
    const float bib = bias_im[atile * 16 + l15];

    // preload the wave's B fragments (G tile column) for all 16 k-steps:
    // reused across all 6 bm-tiles below.
    v2f brf[16], bif[16];
    {
        const float* GrR = Gr + (atile * 16 + l15) * PITCH + kbase;
        const float* GiR = Gi + (atile * 16 + l15) * PITCH + kbase;
#pragma unroll
        for (int kk = 0; kk < 16; ++kk) {
            brf[kk] = *(const v2f*)(GrR + kk * 4);
            bif[kk] = *(const v2f*)(GiR + kk * 4);
        }
    }

    for (int it = 0; it < 6; ++it) {
        int bmtile = (w >> 2) * 6 + it;
        v8f RR = {}, II = {}, RI = {}, IR = {};
        const float* XrR = Xr + (bmtile * 16 + l15) * PITCH + kbase;
        const float* XiR = Xi + (bmtile * 16 + l15) * PITCH + kbase;
#pragma unroll
        for (int kk = 0; kk < 16; ++kk) {
            v2f ar = *(const v2f*)(XrR + kk * 4);
            v2f ai = *(const v2f*)(XiR + kk * 4);
            RR = __builtin_amdgcn_wmma_f32_16x16x4_f32(false, ar, false, brf[kk], (short)0, RR, false, false);
            II = __builtin_amdgcn_wmma_f32_16x16x4_f32(false, ai, false, bif[kk], (short)0, II, false, false);
            RI = __builtin_amdgcn_wmma_f32_16x16x4_f32(false, ar, false, bif[kk], (short)0, RI, false, false);
            IR = __builtin_amdgcn_wmma_f32_16x16x4_f32(false, ai, false, brf[kk], (short)0, IR, false, false);
        }
#pragma unroll
        for (int j = 0; j < 8; ++j) {
            int bm = bmtile * 16 + ((lane < 16) ? j : j + 8);
            int a  = atile * 16 + l15;
            float yr = fmaxf(RR[j] - II[j] + brb, 0.0f);
            float yi = fmaxf(RI[j] + IR[j] + bib, 0.0f);
            spec[(size_t)(bm * 64 + a) * FP + f] = make_float2(yr, yi);
        }
    }
}

// ---------------------------------------------------------------------------
// Kernel 3: inverse real FFT (n=4096) per (bm, a) series -> out [B,T,E].
// ---------------------------------------------------------------------------
__global__ __launch_bounds__(256) void inv_rfft_kernel(
        const float2* __restrict__ spec, float* __restrict__ out)
{
    __shared__ float2 bufA[N2];
    __shared__ float2 bufB[N2];
    const int tid = threadIdx.x;
    const int s   = blockIdx.x;          // bm*64 + a
    const int bm  = s >> 6;
    const int a   = s & 63;
    const int b   = bm / 12;
    const int mb  = bm - b * 12;
    const int eo  = mb * 64 + a;

    const float2* in = spec + (size_t)s * FP;
    for (int i = 0; i < 8; ++i) {
        int n = tid + i * 256;           // 0..2047
        float2 Xk = in[n];
        float2 Xm = in[N2 - n];          // n=0 -> bin 2048 (valid)
        Xm.y = -Xm.y;
        float2 Ev = make_float2(0.5f * (Xk.x + Xm.x), 0.5f * (Xk.y + Xm.y));
        float2 Dv = make_float2(0.5f * (Xk.x - Xm.x), 0.5f * (Xk.y - Xm.y));
        float sn, cs;
        __sincosf(PI_F * (float)n / 2048.0f, &sn, &cs);   // W^{-n}
        float2 Ov = make_float2(cs * Dv.x - sn * Dv.y, cs * Dv.y + sn * Dv.x);
        bufA[n] = make_float2(Ev.x - Ov.y, Ev.y + Ov.x);  // E + i*O
    }

    float2* src = bufA;
    float2* dst = bufB;
    int lm = 0, m = 1;
    for (int l = N2 / 2; l >= 1; l >>= 1) {
        __syncthreads();
        float angScale = PI_F / (float)l;                 // conjugate twiddles
        for (int i = 0; i < 4; ++i) {
            int t = tid + i * 256;
            int k = t & (m - 1);
            int j = t >> lm;
            float2 c0 = src[k + (j << lm)];
            float2 c1 = src[k + (j << lm) + N2 / 2];
            float2 su = make_float2(c0.x + c1.x, c0.y + c1.y);
            float2 di = make_float2(c0.x - c1.x, c0.y - c1.y);
            float sn, cs;
            __sincosf(angScale * (float)j, &sn, &cs);
            dst[k + (j << (lm + 1))]     = su;
            dst[k + (j << (lm + 1)) + m] = make_float2(cs * di.x - sn * di.y,
                                                       cs * di.y + sn * di.x);
        }
        float2* tp = src; src = dst; dst = tp;
        m <<= 1; ++lm;
    }
    __syncthreads();

    float* ob = out + (size_t)b * T_LEN * E_TOT + eo;
    const float sc = 1.0f / 2048.0f;
    for (int i = 0; i < 8; ++i) {
        int n = tid + i * 256;
        float2 z = src[n];
        ob[(size_t)(2 * n    ) * E_TOT] = z.x * sc;
        ob[(size_t)(2 * n + 1) * E_TOT] = z.y * sc;
    }
}

// ---------------------------------------------------------------------------
extern "C" void kernel_launch(void* const* d_in, const int* in_sizes, int n_in,
                              void* d_out, int out_size, void* d_ws, size_t ws_size,
                              hipStream_t stream)
{
    (void)in_sizes; (void)n_in; (void)out_size; (void)ws_size;
    const float* x       = (const float*)d_in[0];
    const float* A_re    = (const float*)d_in[1];
    const float* A_im    = (const float*)d_in[2];
    const float* B_re    = (const float*)d_in[3];
    const float* B_im    = (const float*)d_in[4];
    const float* C_re    = (const float*)d_in[5];
    const float* C_im    = (const float*)d_in[6];
    const float* bias_re = (const float*)d_in[7];
    const float* bias_im = (const float*)d_in[8];

    float2* spec = (float2*)d_ws;       // [192][64][FP] complex, ~202 MB

    // dynamic LDS: 6 planes of 64xPITCH + 2 planes of 192xPITCH + invden
    size_t smem = (size_t)(6 * 64 * PITCH + 2 * 192 * PITCH) * sizeof(float)
                + 64 * sizeof(float2);
    (void)hipFuncSetAttribute((const void*)freq_matmul_kernel,
                              hipFuncAttributeMaxDynamicSharedMemorySize,
                              (int)smem);

    fwd_rfft_kernel<<<NSER, 256, 0, stream>>>(x, spec);
    freq_matmul_kernel<<<NFREQ, 256, smem, stream>>>(spec, A_re, A_im, B_re, B_im,
                                                     C_re, C_im, bias_re, bias_im);
    inv_rfft_kernel<<<NSER, 256, 0, stream>>>(spec, (float*)d_out);
}